// GCN_3layers_72035191489124
// MI455X (gfx1250) — compile-verified
//
#include <hip/hip_runtime.h>
#include <hip/hip_bf16.h>
#include <math.h>

// ---------------------------------------------------------------------------
// GCN 3-layer forward for MI455X (gfx1250).
//   x[100000,512] --GEMM(W1)--> xw[100000,20] --norm-scatter-add--> h1
//   dropout -> GEMM(W2) -> scatter -> dropout -> GEMM(W3,16) -> scatter -> out
// GEMMs use V_WMMA_F32_16X16X4_F32 (fp32 WMMA). Aggregation is edge-parallel
// with f32 L2 atomics (node feature matrices are 8MB -> fully L2 resident).
// LDS weight tiles carry a zero pad column so the second N-tile's B loads are
// unconditional (no EXEC divergence in the WMMA hot loop).
// ---------------------------------------------------------------------------

typedef __attribute__((ext_vector_type(2))) float v2f;
typedef __attribute__((ext_vector_type(8))) float v8f;

#define NFEAT 512
#define WAVES_PER_BLOCK 8

// ---------------- degree / normalization ----------------------------------

__global__ void deg_init_kernel(float* deg, int n) {
  int i = blockIdx.x * blockDim.x + threadIdx.x;
  if (i < n) deg[i] = 1.0f;  // self loop
}

__global__ void deg_count_kernel(const int* __restrict__ dst, float* deg, int e) {
  int i = blockIdx.x * blockDim.x + threadIdx.x;
  if (i < e) atomicAdd(&deg[dst[i]], 1.0f);
}

__global__ void dinv_kernel(float* deg, int n) {
  int i = blockIdx.x * blockDim.x + threadIdx.x;
  if (i < n) deg[i] = rsqrtf(deg[i]);  // deg >= 1 always (self loops)
}

// ---------------- WMMA GEMM: xw = x @ W  (K = 512, Fout = 20) --------------
// One wave -> 16 rows x 20 cols (two 16-wide N tiles; second tile's invalid
// columns read a zeroed pad column -> no divergence).
// A lane layout (16x4 f32 A operand, 2 VGPRs):
//   lane l: m = l&15, kk = (l>>4)*2 ; a = { A[m][k0+kk], A[m][k0+kk+1] }
// B mirrors: b = { B[k0+kk][n], B[k0+kk+1][n] }.
// C/D: vgpr i holds (M = i + (l>>4)*8, N = l&15).

#define LDW1 21  // 20 cols + 1 zero pad column

__global__ __launch_bounds__(256) void gemm1_wmma_kernel(
    const float* __restrict__ x, const float* __restrict__ W,
    float* __restrict__ xw, int nrows) {
  __shared__ float sW[NFEAT * LDW1];  // 42 KB of the 320 KB WGP LDS
  for (int i = threadIdx.x; i < NFEAT * LDW1; i += blockDim.x) {
    int r = i / LDW1;
    int c = i - r * LDW1;
    sW[i] = (c < 20) ? W[r * 20 + c] : 0.0f;
  }
  __syncthreads();

  const int wave = threadIdx.x >> 5;
  const int lane = threadIdx.x & 31;
  const int row_base = (blockIdx.x * WAVES_PER_BLOCK + wave) * 16;
  if (row_base >= nrows) return;  // wave-uniform: EXEC stays all-ones

  const int m  = lane & 15;
  const int kk = (lane >> 4) * 2;
  int row = row_base + m;
  if (row >= nrows) row = nrows - 1;  // load clamp (N is a multiple of 16 anyway)
  const float* __restrict__ xrow = x + (size_t)row * NFEAT;

  const int n1  = 16 + m;
  const int n1c = (n1 < 20) ? n1 : 20;  // clamp to zero pad column

  v8f c0 = {};  // cols 0..15
  v8f c1 = {};  // cols 16..19 (rest hit the zero column)

  for (int k0 = 0; k0 < NFEAT; k0 += 4) {
    const int kb = (k0 + kk) * LDW1;
    v2f a;
    a.x = xrow[k0 + kk];
    a.y = xrow[k0 + kk + 1];
    v2f b0;
    b0.x = sW[kb + m];
    b0.y = sW[kb + LDW1 + m];
    v2f b1;
    b1.x = sW[kb + n1c];
    b1.y = sW[kb + LDW1 + n1c];
    c0 = __builtin_amdgcn_wmma_f32_16x16x4_f32(false, a, false, b0, (short)0, c0, false, false);
    c1 = __builtin_amdgcn_wmma_f32_16x16x4_f32(false, a, false, b1, (short)0, c1, false, false);
  }

  const int mrow = (lane >> 4) * 8;
  const int col  = lane & 15;
  for (int i = 0; i < 8; ++i) {
    size_t r = (size_t)(row_base + mrow + i);
    if (r < (size_t)nrows) {
      xw[r * 20 + col] = c0[i];
      if (col < 4) xw[r * 20 + 16 + col] = c1[i];
    }
  }
}

// ---------------- WMMA GEMM: out = h @ W  (K = 20, Fout = 20 or 16) --------

__global__ __launch_bounds__(256) void gemm_small_wmma_kernel(
    const float* __restrict__ h, const float* __restrict__ W,
    float* __restrict__ out, int nrows, int K, int Fout) {
  __shared__ float sW[20 * 24];  // up to 20 x (20+1), padded stride
  const int ldw = Fout + 1;      // zero pad column at index Fout
  for (int i = threadIdx.x; i < K * ldw; i += blockDim.x) {
    int r = i / ldw;
    int c = i - r * ldw;
    sW[i] = (c < Fout) ? W[r * Fout + c] : 0.0f;
  }
  __syncthreads();

  const int wave = threadIdx.x >> 5;
  const int lane = threadIdx.x & 31;
  const int row_base = (blockIdx.x * WAVES_PER_BLOCK + wave) * 16;
  if (row_base >= nrows) return;

  const int m  = lane & 15;
  const int kk = (lane >> 4) * 2;
  int row = row_base + m;
  if (row >= nrows) row = nrows - 1;
  const float* __restrict__ hrow = h + (size_t)row * K;

  const bool two = (Fout > 16);         // grid-uniform
  const int n1  = 16 + m;
  const int n1c = (n1 < Fout) ? n1 : Fout;  // clamp to zero pad column

  v8f c0 = {};
  v8f c1 = {};

  for (int k0 = 0; k0 < K; k0 += 4) {
    const int kb = (k0 + kk) * ldw;
    v2f a;
    a.x = hrow[k0 + kk];
    a.y = hrow[k0 + kk + 1];
    v2f b0;
    b0.x = sW[kb + m];
    b0.y = sW[kb + ldw + m];
    c0 = __builtin_amdgcn_wmma_f32_16x16x4_f32(false, a, false, b0, (short)0, c0, false, false);
    if (two) {
      v2f b1;
      b1.x = sW[kb + n1c];
      b1.y = sW[kb + ldw + n1c];
      c1 = __builtin_amdgcn_wmma_f32_16x16x4_f32(false, a, false, b1, (short)0, c1, false, false);
    }
  }

  const int mrow = (lane >> 4) * 8;
  const int col  = lane & 15;
  for (int i = 0; i < 8; ++i) {
    size_t r = (size_t)(row_base + mrow + i);
    if (r < (size_t)nrows) {
      out[r * Fout + col] = c0[i];
      if (two && (16 + col) < Fout) out[r * Fout + 16 + col] = c1[i];
    }
  }
}

// ---------------- aggregation --------------------------------------------
// init: out[i,f] = bias[f] + dinv[i]^2 * xw[i,f]   (self-loop term folded in)

__global__ void init_agg_kernel(const float* __restrict__ xw,
                                const float* __restrict__ dinv,
                                const float* __restrict__ bias,
                                float* __restrict__ out, int n, int F) {
  int idx = blockIdx.x * blockDim.x + threadIdx.x;
  if (idx >= n * F) return;
  int i = idx / F;
  int f = idx - i * F;
  float di = dinv[i];
  out[idx] = bias[f] + di * di * xw[idx];
}

// edge scatter: out[dst] += dinv[src]*dinv[dst] * xw[src]  (F = 20 or 16;
// row stride in bytes is 80 / 64 -> float4 aligned)

__global__ void agg_edges_kernel(const float* __restrict__ xw,
                                 const int* __restrict__ src,
                                 const int* __restrict__ dst,
                                 const float* __restrict__ dinv,
                                 float* __restrict__ out, int e, int F4, int F) {
  int idx = blockIdx.x * blockDim.x + threadIdx.x;
  if (idx >= e) return;
  int s = src[idx];
  int d = dst[idx];
  float nrm = dinv[s] * dinv[d];
  const float4* __restrict__ xr = reinterpret_cast<const float4*>(xw + (size_t)s * F);
  float* orow = out + (size_t)d * F;
#pragma unroll 5
  for (int q = 0; q < F4; ++q) {
    float4 v = xr[q];
    atomicAdd(orow + 4 * q + 0, nrm * v.x);
    atomicAdd(orow + 4 * q + 1, nrm * v.y);
    atomicAdd(orow + 4 * q + 2, nrm * v.z);
    atomicAdd(orow + 4 * q + 3, nrm * v.w);
  }
}

// ---------------- dropout (deterministic counter-based hash) ---------------

__device__ __forceinline__ unsigned wang_hash(unsigned s) {
  s = (s ^ 61u) ^ (s >> 16);
  s *= 9u;
  s ^= s >> 4;
  s *= 0x27d4eb2du;
  s ^= s >> 15;
  return s;
}

__global__ void dropout_kernel(float* __restrict__ h, int n, unsigned seed) {
  int idx = blockIdx.x * blockDim.x + threadIdx.x;
  if (idx >= n) return;
  unsigned r = wang_hash((unsigned)idx * 0x9E3779B9u + seed);
  h[idx] = (r & 0x80000000u) ? h[idx] * 2.0f : 0.0f;  // p = 0.5, scale 1/(1-p)
}

// ---------------------------------------------------------------------------

extern "C" void kernel_launch(void* const* d_in, const int* in_sizes, int n_in,
                              void* d_out, int out_size, void* d_ws, size_t ws_size,
                              hipStream_t stream) {
  const float* x  = (const float*)d_in[0];
  const int* edge = (const int*)d_in[1];
  const float* W1 = (const float*)d_in[2];
  const float* b1 = (const float*)d_in[3];
  const float* W2 = (const float*)d_in[4];
  const float* b2 = (const float*)d_in[5];
  const float* W3 = (const float*)d_in[6];
  const float* b3 = (const float*)d_in[7];
  float* out = (float*)d_out;

  const int N = in_sizes[0] / NFEAT;   // 100000
  const int E = in_sizes[1] / 2;       // 3200000
  const int* src = edge;
  const int* dst = edge + E;

  // workspace: dinv[N] | bufA[N*20] | bufB[N*20]   (all 16B aligned)
  char* ws = (char*)d_ws;
  float* dinv = (float*)ws;
  float* bufA = (float*)(ws + (size_t)N * sizeof(float));
  float* bufB = bufA + (size_t)N * 20;

  const int T = 256;
  const int gN  = (N + T - 1) / T;
  const int gE  = (E + T - 1) / T;
  const int g20 = (N * 20 + T - 1) / T;
  const int g16 = (N * 16 + T - 1) / T;
  const int tiles = (N + 15) / 16;
  const int gGemm = (tiles + WAVES_PER_BLOCK - 1) / WAVES_PER_BLOCK;

  // graph normalization
  deg_init_kernel<<<gN, T, 0, stream>>>(dinv, N);
  deg_count_kernel<<<gE, T, 0, stream>>>(dst, dinv, E);
  dinv_kernel<<<gN, T, 0, stream>>>(dinv, N);

  // ---- layer 1: x @ W1 -> aggregate -> dropout
  gemm1_wmma_kernel<<<gGemm, T, 0, stream>>>(x, W1, bufA, N);
  init_agg_kernel<<<g20, T, 0, stream>>>(bufA, dinv, b1, bufB, N, 20);
  agg_edges_kernel<<<gE, T, 0, stream>>>(bufA, src, dst, dinv, bufB, E, 5, 20);
  dropout_kernel<<<g20, T, 0, stream>>>(bufB, N * 20, 0x1234567u);

  // ---- layer 2: h1 @ W2 -> aggregate -> dropout
  gemm_small_wmma_kernel<<<gGemm, T, 0, stream>>>(bufB, W2, bufA, N, 20, 20);
  init_agg_kernel<<<g20, T, 0, stream>>>(bufA, dinv, b2, bufB, N, 20);
  agg_edges_kernel<<<gE, T, 0, stream>>>(bufA, src, dst, dinv, bufB, E, 5, 20);
  dropout_kernel<<<g20, T, 0, stream>>>(bufB, N * 20, 0x89ABCDEFu);

  // ---- layer 3: h2 @ W3 -> aggregate (into d_out, F = 16)
  gemm_small_wmma_kernel<<<gGemm, T, 0, stream>>>(bufB, W3, bufA, N, 20, 16);
  init_agg_kernel<<<g16, T, 0, stream>>>(bufA, dinv, b3, out, N, 16);
  agg_edges_kernel<<<gE, T, 0, stream>>>(bufA, src, dst, dinv, out, E, 4, 16);
}